// CrossAttention_37056977830404
// MI455X (gfx1250) — compile-verified
//
#include <hip/hip_runtime.h>
#include <hip/hip_bf16.h>

typedef __attribute__((ext_vector_type(2))) float v2f;
typedef __attribute__((ext_vector_type(8))) float v8f;

#define NB 8
#define NH 8
#define HD 64
#define C1 512
#define C2 256
#define N1 3136   /* 56*56 */
#define N2 12544  /* 112*112 */
#define ROWS (NB * N1) /* 25088 */

// ---------------------------------------------------------------------------
// CDNA5 async global->LDS copy support (ASYNCcnt path). The builtin's
// signature (from the compiler diagnostic) is:
//   void __builtin_amdgcn_global_load_async_to_lds_b128(
//       int __vector_size(16)* addrspace(1) src,
//       int __vector_size(16)* addrspace(3) dst, imm int offset, imm int cpol)
// Guarded so the host pass (no amdgcn builtins) still parses the fallback.
// ---------------------------------------------------------------------------
#if defined(__AMDGCN__) && __has_builtin(__builtin_amdgcn_global_load_async_to_lds_b128)
#define USE_ASYNC_LDS 1
typedef int v4i_b128 __attribute__((__vector_size__(16)));
typedef __attribute__((address_space(1))) v4i_b128 as1_v4i;
typedef __attribute__((address_space(3))) v4i_b128 as3_v4i;
#if __has_builtin(__builtin_amdgcn_s_wait_asynccnt)
#define ASYNC_WAIT(n) __builtin_amdgcn_s_wait_asynccnt(n)
#else
#define ASYNC_WAIT(n) asm volatile("s_wait_asynccnt %0" :: "n"(n) : "memory")
#endif
#else
#define USE_ASYNC_LDS 0
#endif

// ---------------------------------------------------------------------------
// Generic fp32 WMMA GEMM: C[M,N] = A[M,K] @ B[N,K]^T (+ bias[N])
// block = 256 threads = 8 waves; tile 64x64; wave -> 16(M) x 32(N).
// Tiles staged in LDS; with async support: double-buffered LDS fills via
// GLOBAL_LOAD_ASYNC_TO_LDS_B128 overlapped with WMMA compute.
// ---------------------------------------------------------------------------
#define TM 64
#define TN 64
#define TK 32
#define LDT 34  /* TK + 2 pad: breaks ds bank conflicts for row-strided frags */

__global__ __launch_bounds__(256)
void gemm_nt_f32_wmma(const float* __restrict__ A, const float* __restrict__ B,
                      const float* __restrict__ bias, float* __restrict__ C,
                      int M, int N, int K)
{
    const int tid  = threadIdx.x;
    const int lane = tid & 31;
    const int wave = tid >> 5;      // 0..7
    const int wm   = wave & 3;      // M sub-tile 0..3
    const int wn   = wave >> 2;     // N sub-tile 0..1

    const int nTilesN = N / TN;
    const int bm = blockIdx.x / nTilesN;
    const int bn = blockIdx.x % nTilesN;

    const float* Ab = A + (size_t)bm * TM * K;
    const float* Bb = B + (size_t)bn * TN * K;

    v8f acc0 = {};
    v8f acc1 = {};

    const int ll   = lane & 15;
    const int koff = (lane >> 4) * 2;   // 0 or 2

    // cooperative load mapping: 64x32 floats = 512 float4; 2 per thread
    const int r0 = tid >> 3;            // rows for idx = tid, tid+256
    const int c0 = (tid & 7) * 4;
    const int r1 = (tid + 256) >> 3;
    const int c1 = c0;

#if USE_ASYNC_LDS
    __shared__ float As[2][TM * LDT];
    __shared__ float Bs[2][TN * LDT];

    // issue one tile's async fills (4 x B128 per thread -> ASYNCcnt += 4)
    auto issue_tile = [&](int k0, int buf) {
        __builtin_amdgcn_global_load_async_to_lds_b128(
            (as1_v4i*)(Ab + (size_t)r0 * K + k0 + c0),
            (as3_v4i*)&As[buf][r0 * LDT + c0], 0, 0);
        __builtin_amdgcn_global_load_async_to_lds_b128(
            (as1_v4i*)(Bb + (size_t)r0 * K + k0 + c0),
            (as3_v4i*)&Bs[buf][r0 * LDT + c0], 0, 0);
        __builtin_amdgcn_global_load_async_to_lds_b128(
            (as1_v4i*)(Ab + (size_t)r1 * K + k0 + c1),
            (as3_v4i*)&As[buf][r1 * LDT + c1], 0, 0);
        __builtin_amdgcn_global_load_async_to_lds_b128(
            (as1_v4i*)(Bb + (size_t)r1 * K + k0 + c1),
            (as3_v4i*)&Bs[buf][r1 * LDT + c1], 0, 0);
    };

    const int nk = K / TK;
    issue_tile(0, 0);
    for (int t = 0; t < nk; ++t) {
        const int buf = t & 1;
        if (t + 1 < nk) {
            issue_tile((t + 1) * TK, (t + 1) & 1);  // overlap next fill
            ASYNC_WAIT(4);                           // tile t done (in-order)
        } else {
            ASYNC_WAIT(0);
        }
        __syncthreads();
        const float* Asb = As[buf];
        const float* Bsb = Bs[buf];
#pragma unroll
        for (int kk = 0; kk < TK; kk += 4) {
            v2f a  = *(const v2f*)(&Asb[(wm * 16      + ll) * LDT + kk + koff]);
            v2f b0 = *(const v2f*)(&Bsb[(wn * 32      + ll) * LDT + kk + koff]);
            v2f b1 = *(const v2f*)(&Bsb[(wn * 32 + 16 + ll) * LDT + kk + koff]);
            acc0 = __builtin_amdgcn_wmma_f32_16x16x4_f32(false, a, false, b0,
                                                         (short)0, acc0, false, false);
            acc1 = __builtin_amdgcn_wmma_f32_16x16x4_f32(false, a, false, b1,
                                                         (short)0, acc1, false, false);
        }
        __syncthreads();   // buffer safe to refill next iteration
    }
#else
    __shared__ float As[TM * LDT];
    __shared__ float Bs[TN * LDT];

    for (int k0 = 0; k0 < K; k0 += TK) {
        {
            float4 a4 = *(const float4*)(Ab + (size_t)r0 * K + k0 + c0);
            As[r0 * LDT + c0 + 0] = a4.x;
            As[r0 * LDT + c0 + 1] = a4.y;
            As[r0 * LDT + c0 + 2] = a4.z;
            As[r0 * LDT + c0 + 3] = a4.w;
            float4 b4 = *(const float4*)(Bb + (size_t)r0 * K + k0 + c0);
            Bs[r0 * LDT + c0 + 0] = b4.x;
            Bs[r0 * LDT + c0 + 1] = b4.y;
            Bs[r0 * LDT + c0 + 2] = b4.z;
            Bs[r0 * LDT + c0 + 3] = b4.w;
            float4 a5 = *(const float4*)(Ab + (size_t)r1 * K + k0 + c1);
            As[r1 * LDT + c1 + 0] = a5.x;
            As[r1 * LDT + c1 + 1] = a5.y;
            As[r1 * LDT + c1 + 2] = a5.z;
            As[r1 * LDT + c1 + 3] = a5.w;
            float4 b5 = *(const float4*)(Bb + (size_t)r1 * K + k0 + c1);
            Bs[r1 * LDT + c1 + 0] = b5.x;
            Bs[r1 * LDT + c1 + 1] = b5.y;
            Bs[r1 * LDT + c1 + 2] = b5.z;
            Bs[r1 * LDT + c1 + 3] = b5.w;
        }
        __syncthreads();
#pragma unroll
        for (int kk = 0; kk < TK; kk += 4) {
            v2f a  = *(const v2f*)(&As[(wm * 16      + ll) * LDT + kk + koff]);
            v2f b0 = *(const v2f*)(&Bs[(wn * 32      + ll) * LDT + kk + koff]);
            v2f b1 = *(const v2f*)(&Bs[(wn * 32 + 16 + ll) * LDT + kk + koff]);
            acc0 = __builtin_amdgcn_wmma_f32_16x16x4_f32(false, a, false, b0,
                                                         (short)0, acc0, false, false);
            acc1 = __builtin_amdgcn_wmma_f32_16x16x4_f32(false, a, false, b1,
                                                         (short)0, acc1, false, false);
        }
        __syncthreads();
    }
#endif

    const int hi   = (lane >> 4) * 8;
    const int col0 = bn * TN + wn * 32 + ll;
    const float bv0 = bias ? bias[col0]      : 0.0f;
    const float bv1 = bias ? bias[col0 + 16] : 0.0f;
#pragma unroll
    for (int i = 0; i < 8; ++i) {
        int row = bm * TM + wm * 16 + hi + i;
        C[(size_t)row * N + col0]      = acc0[i] + bv0;
        C[(size_t)row * N + col0 + 16] = acc1[i] + bv1;
    }
}

// ---------------------------------------------------------------------------
// 2x2 average pool: y (B, 112*112, 256) -> pooled (B*3136, 256)
// ---------------------------------------------------------------------------
__global__ __launch_bounds__(256)
void pool2x2_kernel(const float* __restrict__ Y, float* __restrict__ P)
{
    int idx = blockIdx.x * 256 + threadIdx.x;   // ROWS*C2 total
    int c = idx & 255;
    int r = idx >> 8;
    int b  = r / N1;
    int pq = r - b * N1;
    int hp = pq / 56, wp = pq - hp * 56;
    int h2 = hp * 2, w2 = wp * 2;
    size_t base = (size_t)b * N2 * C2;
    float s = Y[base + (size_t)(h2 * 112 + w2)       * C2 + c]
            + Y[base + (size_t)(h2 * 112 + w2 + 1)   * C2 + c]
            + Y[base + (size_t)((h2 + 1) * 112 + w2)     * C2 + c]
            + Y[base + (size_t)((h2 + 1) * 112 + w2 + 1) * C2 + c];
    P[(size_t)r * C2 + c] = 0.25f * s;
}

// ---------------------------------------------------------------------------
// Per-row LayerNorm over 256 channels + exact GELU, in place.
// 8 waves/block, one row per wave, wave-shuffle reduction (wave32).
// ---------------------------------------------------------------------------
__global__ __launch_bounds__(256)
void ln_gelu_kernel(float* __restrict__ X, const float* __restrict__ gn,
                    const float* __restrict__ bn, int nrows)
{
    int wave = threadIdx.x >> 5;
    int lane = threadIdx.x & 31;
    int row  = blockIdx.x * 8 + wave;
    if (row >= nrows) return;
    float* xr = X + (size_t)row * C2;

    float v[8];
    float s = 0.0f, s2 = 0.0f;
#pragma unroll
    for (int i = 0; i < 8; ++i) {
        v[i] = xr[lane + i * 32];
        s  += v[i];
        s2 += v[i] * v[i];
    }
#pragma unroll
    for (int off = 16; off > 0; off >>= 1) {
        s  += __shfl_xor(s,  off, 32);
        s2 += __shfl_xor(s2, off, 32);
    }
    float mean = s * (1.0f / 256.0f);
    float var  = s2 * (1.0f / 256.0f) - mean * mean;
    float rstd = rsqrtf(var + 1e-5f);
#pragma unroll
    for (int i = 0; i < 8; ++i) {
        int c = lane + i * 32;
        float t = (v[i] - mean) * rstd * gn[c] + bn[c];
        xr[c] = 0.5f * t * (1.0f + erff(t * 0.70710678118654752f));
    }
}

// ---------------------------------------------------------------------------
// Windowed attention, fp32 WMMA. One wave per (b, head, window).
// Q: (B*N1, 512) laid [head*64+d]; KV: (B*N1, 1024) K then V.
// Output written window-reversed into O: (B*N1, 512).
// ---------------------------------------------------------------------------
#define LDQ 66  /* 64 + 2 pad */
#define LDV 54  /* 52 + 2 pad */

__global__ __launch_bounds__(32)
void win_attn_kernel(const float* __restrict__ Q, const float* __restrict__ KV,
                     float* __restrict__ O)
{
    __shared__ float Qs[64 * LDQ];
    __shared__ float Ks[64 * LDQ];
    __shared__ float Ss[64 * LDQ];
    __shared__ float Vt[64 * LDV];   // transposed: [d][token]

    const int lane = threadIdx.x;
    const int wi = blockIdx.x & 63;
    const int bh = blockIdx.x >> 6;
    const int b  = bh >> 3;
    const int h  = bh & 7;
    const int wy = wi >> 3, wx = wi & 7;

    const int d = lane * 2;   // 0..62

    // ---- stage Q, K (row-major) and V (transposed) tiles ----
    for (int t = 0; t < 49; ++t) {
        int ty = t / 7, tx = t - ty * 7;
        int p  = (wy * 7 + ty) * 56 + (wx * 7 + tx);
        size_t row = (size_t)(b * N1 + p);
        float2 qv = *(const float2*)(Q  + row * C1   + h * HD + d);
        Qs[t * LDQ + d]     = qv.x;
        Qs[t * LDQ + d + 1] = qv.y;
        float2 kv = *(const float2*)(KV + row * 1024 + h * HD + d);
        Ks[t * LDQ + d]     = kv.x;
        Ks[t * LDQ + d + 1] = kv.y;
        float2 vv = *(const float2*)(KV + row * 1024 + 512 + h * HD + d);
        Vt[(d)     * LDV + t] = vv.x;
        Vt[(d + 1) * LDV + t] = vv.y;
    }
    // zero padding: Q/K rows 49..63, V columns 49..51
    for (int t = 49; t < 64; ++t) {
        Qs[t * LDQ + d] = 0.0f; Qs[t * LDQ + d + 1] = 0.0f;
        Ks[t * LDQ + d] = 0.0f; Ks[t * LDQ + d + 1] = 0.0f;
    }
#pragma unroll
    for (int c = 49; c < 52; ++c) {
        Vt[(d)     * LDV + c] = 0.0f;
        Vt[(d + 1) * LDV + c] = 0.0f;
    }
    __syncthreads();   // single-wave block: S_NOP; LDS same-wave ordering suffices

    const int ll   = lane & 15;
    const int koff = (lane >> 4) * 2;
    const int hi   = (lane >> 4) * 8;

    // ---- S = Q @ K^T ----
#pragma unroll
    for (int mt = 0; mt < 4; ++mt) {
#pragma unroll
        for (int nt = 0; nt < 4; ++nt) {
            v8f acc = {};
#pragma unroll
            for (int kk = 0; kk < 64; kk += 4) {
                v2f a  = *(const v2f*)(&Qs[(mt * 16 + ll) * LDQ + kk + koff]);
                v2f bb = *(const v2f*)(&Ks[(nt * 16 + ll) * LDQ + kk + koff]);
                acc = __builtin_amdgcn_wmma_f32_16x16x4_f32(false, a, false, bb,
                                                            (short)0, acc, false, false);
            }
#pragma unroll
            for (int i = 0; i < 8; ++i)
                Ss[(mt * 16 + hi + i) * LDQ + nt * 16 + ll] = acc[i];
        }
    }

    // ---- softmax over valid 49 columns (scale = hd^-0.5 = 0.125) ----
    const float sc = 0.125f;
    for (int r = lane; r < 49; r += 32) {
        float mx = -1e30f;
        for (int j = 0; j < 49; ++j) mx = fmaxf(mx, Ss[r * LDQ + j]);
        mx *= sc;
        float sum = 0.0f;
        for (int j = 0; j < 49; ++j) {
            float e = expf(Ss[r * LDQ + j] * sc - mx);
            Ss[r * LDQ + j] = e;
            sum += e;
        }
        float inv = 1.0f / sum;
        for (int j = 0; j < 49; ++j) Ss[r * LDQ + j] *= inv;
        Ss[r * LDQ + 49] = 0.0f;
        Ss[r * LDQ + 50] = 0.0f;
        Ss[r * LDQ + 51] = 0.0f;
    }

    // ---- O = P @ V  (K dim padded to 52) ----
#pragma unroll
    for (int mt = 0; mt < 4; ++mt) {
#pragma unroll
        for (int nt = 0; nt < 4; ++nt) {
            v8f acc = {};
#pragma unroll
            for (int kk = 0; kk < 52; kk += 4) {
                v2f a  = *(const v2f*)(&Ss[(mt * 16 + ll) * LDQ + kk + koff]);
                v2f bb = *(const v2f*)(&Vt[(nt * 16 + ll) * LDV + kk + koff]);
                acc = __builtin_amdgcn_wmma_f32_16x16x4_f32(false, a, false, bb,
                                                            (short)0, acc, false, false);
            }
#pragma unroll
            for (int i = 0; i < 8; ++i) {
                int r = mt * 16 + hi + i;
                if (r < 49) {   // window reverse on store
                    int ty = r / 7, tx = r - ty * 7;
                    int p  = (wy * 7 + ty) * 56 + (wx * 7 + tx);
                    O[(size_t)(b * N1 + p) * C1 + h * HD + nt * 16 + ll] = acc[i];
                }
            }
        }
    }
}

// ---------------------------------------------------------------------------
// Host launcher
// ---------------------------------------------------------------------------
extern "C" void kernel_launch(void* const* d_in, const int* in_sizes, int n_in,
                              void* d_out, int out_size, void* d_ws, size_t ws_size,
                              hipStream_t stream)
{
    const float* x     = (const float*)d_in[0];
    const float* y     = (const float*)d_in[1];
    const float* Wq    = (const float*)d_in[2];
    const float* Wkv   = (const float*)d_in[3];
    const float* Wproj = (const float*)d_in[4];
    const float* bproj = (const float*)d_in[5];
    const float* Wsr   = (const float*)d_in[6];
    const float* bsr   = (const float*)d_in[7];
    const float* gn    = (const float*)d_in[8];
    const float* bn    = (const float*)d_in[9];

    float* ws = (float*)d_ws;
    // layout (floats): [pooled 25088*256][srout 25088*256][Q 25088*512][KV 25088*1024]
    // attention output reuses [pooled+srout] = exactly 25088*512 floats.
    float* pooled = ws;
    float* srout  = ws + (size_t)ROWS * C2;
    float* attn   = ws;                               // overwrites pooled+srout (dead)
    float* Qbuf   = ws + (size_t)ROWS * C2 * 2;
    float* KVbuf  = Qbuf + (size_t)ROWS * C1;
    float* out    = (float*)d_out;

    // 1) Q = x @ Wq^T
    gemm_nt_f32_wmma<<<(ROWS / TM) * (C1 / TN), 256, 0, stream>>>(
        x, Wq, nullptr, Qbuf, ROWS, C1, C1);

    // 2) pooled = avgpool2x2(y)
    pool2x2_kernel<<<(ROWS * C2) / 256, 256, 0, stream>>>(y, pooled);

    // 3) srout = pooled @ Wsr^T + bsr
    gemm_nt_f32_wmma<<<(ROWS / TM) * (C2 / TN), 256, 0, stream>>>(
        pooled, Wsr, bsr, srout, ROWS, C2, C2);

    // 4) layernorm + gelu (in place)
    ln_gelu_kernel<<<ROWS / 8, 256, 0, stream>>>(srout, gn, bn, ROWS);

    // 5) KV = srout @ Wkv^T
    gemm_nt_f32_wmma<<<(ROWS / TM) * (1024 / TN), 256, 0, stream>>>(
        srout, Wkv, nullptr, KVbuf, ROWS, 1024, C2);

    // 6) windowed attention (+ window reverse)
    win_attn_kernel<<<NB * NH * 64, 32, 0, stream>>>(Qbuf, KVbuf, attn);

    // 7) out = attn @ Wproj^T + bproj
    gemm_nt_f32_wmma<<<(ROWS / TM) * (C1 / TN), 256, 0, stream>>>(
        attn, Wproj, bproj, out, ROWS, C1, C1);
}